// knn_feature_11733850653059
// MI455X (gfx1250) — compile-verified
//
#include <hip/hip_runtime.h>

// ---------------- problem constants ----------------
#define BK   4      // batches
#define CC   128    // channels
#define NN   4096   // points
#define OUTC 256    // output channels
#define KNN  10     // neighbors
#define NTILES (NN / 16)     // 256 column tiles
#define WAVES 4
#define ROWS_PER_WG 64       // 4 waves x 16 rows
#define BSTRIDE 132          // padded LDS row stride (floats) -> conflict-free b64 reads

typedef float v2f __attribute__((ext_vector_type(2)));
typedef float v4f __attribute__((ext_vector_type(4)));
typedef float v8f __attribute__((ext_vector_type(8)));

#if defined(__gfx1250__) && __has_builtin(__builtin_amdgcn_wmma_f32_16x16x4_f32)
#define HAS_WMMA_F32 1
#else
#define HAS_WMMA_F32 0
#endif

// ---------------- kernel 1: column squared norms ----------------
__global__ void xx_kernel(const float* __restrict__ x, float* __restrict__ xx) {
  int i = blockIdx.x * blockDim.x + threadIdx.x;   // 0 .. B*N-1
  int b = i >> 12;
  int n = i & (NN - 1);
  const float* xb = x + (size_t)b * CC * NN + n;
  float s = 0.0f;
#pragma unroll 8
  for (int c = 0; c < CC; ++c) {
    float v = xb[(size_t)c * NN];
    s += v * v;
  }
  xx[i] = s;
}

// ---------------- top-k insertion (register resident, fully unrolled) ------
__device__ __forceinline__ void topk_insert(float (&tv)[KNN], int (&ti)[KNN],
                                            float& wmin, float d, int m) {
  if (d < wmin) return;  // early reject (keeps == for tie-break path)
  int   wslot = 0;
  float wv = tv[0];
  int   wi = ti[0];
#pragma unroll
  for (int q = 1; q < KNN; ++q) {
    bool worse = (tv[q] < wv) || ((tv[q] == wv) && (ti[q] > wi));
    if (worse) { wv = tv[q]; wi = ti[q]; wslot = q; }
  }
  bool better = (d > wv) || ((d == wv) && (m < wi));
  if (better) {
#pragma unroll
    for (int q = 0; q < KNN; ++q)
      if (q == wslot) { tv[q] = d; ti[q] = m; }
    float nw = tv[0];
#pragma unroll
    for (int q = 1; q < KNN; ++q) nw = fminf(nw, tv[q]);
    wmin = nw;
  }
}

// ---------------- kernel 2: fused distance WMMA + top-K ----------------
// One workgroup = 64 query rows (wave w -> rows n0+16w..n0+16w+15).
// All 4 waves sweep the same 16-wide m-tile; the 16x128 B tile is staged
// TRANSPOSED into LDS ([m][c], padded stride) once per iteration and shared,
// with double buffering and register-pipelined global loads.
// Key = 2*S[n][m] - xx[m] (row-constant terms dropped; ordering invariant).
__global__ __launch_bounds__(128) void knn_topk_kernel(
    const float* __restrict__ x, const float* __restrict__ xx,
    int* __restrict__ idx) {
  int b    = blockIdx.x / (NN / ROWS_PER_WG);
  int n0   = (blockIdx.x % (NN / ROWS_PER_WG)) * ROWS_PER_WG;
  int tid  = threadIdx.x;
  int w    = tid >> 5;
  int lane = tid & 31;
  int ml   = lane & 15;
  int khalf = (lane >> 4) << 1;  // 0 or 2 (K sub-pair selected by lane half)
  int n0w  = n0 + w * 16;        // this wave's row tile

  const float* xb  = x  + (size_t)b * CC * NN;
  const float* xxb = xx + (size_t)b * NN;

  __shared__ float bt[2][16 * BSTRIDE];   // staged B tile, [m][c] transposed
  __shared__ float tileS[WAVES][16 * 17]; // per-wave pd tile for row scan

  // Preload A tile (rows n0w..n0w+15, all 128 channels): 32 chunks of 2 VGPRs.
  // A layout (16x4 f32): lane<16 -> M=lane holds K=0,1; lane>=16 -> K=2,3.
  v2f a[32];
#pragma unroll
  for (int kc = 0; kc < 32; ++kc) {
    int c = 4 * kc + khalf;
    a[kc].x = xb[(size_t)c * NN + n0w + ml];
    a[kc].y = xb[(size_t)(c + 1) * NN + n0w + ml];
  }

  float tv[KNN];
  int   ti[KNN];
#pragma unroll
  for (int q = 0; q < KNN; ++q) { tv[q] = -3.0e38f; ti[q] = 0x7fffffff; }
  float wmin = -3.0e38f;

  // ---- staging: thread tid owns channel row c=tid; 16 columns per tile ----
  const float* xstage = xb + (size_t)tid * NN;
  v4f r[4];

#define LOAD_TILE(mt)                                              \
  do {                                                             \
    const v4f* p_ = (const v4f*)(xstage + (size_t)(mt) * 16);      \
    r[0] = p_[0]; r[1] = p_[1]; r[2] = p_[2]; r[3] = p_[3];        \
  } while (0)

#define STORE_TILE(bufid)                                          \
  do {                                                             \
    _Pragma("unroll")                                              \
    for (int j_ = 0; j_ < 4; ++j_) {                               \
      _Pragma("unroll")                                            \
      for (int e_ = 0; e_ < 4; ++e_)                               \
        bt[bufid][(4 * j_ + e_) * BSTRIDE + tid] = r[j_][e_];      \
    }                                                              \
  } while (0)

  LOAD_TILE(0);
  STORE_TILE(0);
  LOAD_TILE(1);
  __syncthreads();

  for (int it = 0; it < NTILES; ++it) {
    int cur = it & 1;
    int m0  = it * 16;

    v8f acc = {0.f, 0.f, 0.f, 0.f, 0.f, 0.f, 0.f, 0.f};
#if HAS_WMMA_F32
#pragma unroll
    for (int kc = 0; kc < 32; ++kc) {
      int c = 4 * kc + khalf;
      v2f bv = *(const v2f*)&bt[cur][ml * BSTRIDE + c];  // ds_load_b64
      acc = __builtin_amdgcn_wmma_f32_16x16x4_f32(
          false, a[kc], false, bv, (short)0, acc, false, false);
    }
#else
    {
      int mhf = (lane >> 4) * 8;
#pragma unroll 4
      for (int c = 0; c < CC; ++c) {
        float bvv = bt[cur][ml * BSTRIDE + c];
#pragma unroll
        for (int v = 0; v < 8; ++v)
          acc[v] += xb[(size_t)c * NN + n0w + v + mhf] * bvv;
      }
    }
#endif

    // D layout: lanes 0-15 hold M=v, lanes 16-31 hold M=v+8; col = lane&15.
    float xxm = xxb[m0 + ml];
    int   mh  = (lane >> 4) * 8;
#pragma unroll
    for (int v = 0; v < 8; ++v)
      tileS[w][(v + mh) * 17 + ml] = 2.0f * acc[v] - xxm;

    // wave-local LDS sync (per-wave tile; wave executes in lockstep)
    __asm__ volatile("s_wait_dscnt 0" ::: "memory");

    if (lane < 16) {
      int rrow = lane;  // row owned by this lane
#pragma unroll
      for (int j = 0; j < 16; ++j) {
        float d = tileS[w][rrow * 17 + j];
        topk_insert(tv, ti, wmin, d, m0 + j);
      }
    }

    // stage next tile; prefetch the one after into registers
    if (it + 1 < NTILES) {
      STORE_TILE(1 - cur);
      if (it + 2 < NTILES) LOAD_TILE(it + 2);
    }
    __syncthreads();
  }

  // Each wave scanned all candidates for its own 16 rows: no merge needed.
  if (lane < 16) {
    int* op = idx + ((size_t)b * NN + n0w + lane) * KNN;
#pragma unroll
    for (int q = 0; q < KNN; ++q) op[q] = ti[q];
  }
#undef LOAD_TILE
#undef STORE_TILE
}

// ---------------- kernel 3: combined weights Wc = [W1 ; W2-W1] -------------
__global__ void wc_kernel(const float* __restrict__ W, float* __restrict__ wc) {
  int o = blockIdx.x;
  int r = threadIdx.x;
  float v = W[(size_t)o * (2 * CC) + r];
  if (r >= CC) v -= W[(size_t)o * (2 * CC) + (r - CC)];
  wc[(size_t)o * (2 * CC) + r] = v;
}

// ---------------- kernel 4: gather-mean into stacked g = [f ; x] -----------
// grid: (B*CC, NN/256), 256 threads over n -> coalesced writes; gathers hit L2.
__global__ void gather_kernel(const float* __restrict__ x,
                              const int* __restrict__ idx,
                              float* __restrict__ g) {
  int bc = blockIdx.x;
  int b  = bc >> 7;
  int c  = bc & (CC - 1);
  int n  = blockIdx.y * 256 + threadIdx.x;
  const float* xrow = x + ((size_t)b * CC + c) * NN;
  const int*   ip   = idx + ((size_t)b * NN + n) * KNN;
  float s = 0.0f;
#pragma unroll
  for (int k = 0; k < KNN; ++k) s += xrow[ip[k]];
  float* gb = g + (size_t)b * (2 * CC) * NN;
  gb[(size_t)c * NN + n]        = s * (1.0f / (float)KNN);
  gb[(size_t)(c + CC) * NN + n] = xrow[n];
}

// ---------------- kernel 5: output GEMM (Wc x g) + bias, WMMA --------------
// out[b,o,n] tile 16x16, K-loop over 256 in steps of 4 (64 WMMAs).
// grid: B*16*256/4 blocks of 128 threads (4 waves, one tile each).
__global__ __launch_bounds__(128) void out_gemm_kernel(
    const float* __restrict__ wc, const float* __restrict__ g,
    const float* __restrict__ bias, float* __restrict__ out) {
  int t    = blockIdx.x * 4 + (threadIdx.x >> 5);  // tile id
  int b    = t >> 12;                               // / (16*256)
  int rem  = t & 4095;
  int o0   = (rem >> 8) << 4;
  int n0   = (rem & 255) << 4;
  int lane = threadIdx.x & 31;
  int ml   = lane & 15;
  int khalf = (lane >> 4) << 1;

  const float* gb = g + (size_t)b * (2 * CC) * NN;
  v8f acc = {0.f, 0.f, 0.f, 0.f, 0.f, 0.f, 0.f, 0.f};

#if HAS_WMMA_F32
#pragma unroll
  for (int kc = 0; kc < 64; ++kc) {
    int c = 4 * kc + khalf;
    v2f av = *(const v2f*)(wc + (size_t)(o0 + ml) * (2 * CC) + c);
    v2f bv;
    bv.x = gb[(size_t)c * NN + n0 + ml];
    bv.y = gb[(size_t)(c + 1) * NN + n0 + ml];
    acc = __builtin_amdgcn_wmma_f32_16x16x4_f32(
        false, av, false, bv, (short)0, acc, false, false);
  }
#else
  {
    int mhf = (lane >> 4) * 8;
#pragma unroll 4
    for (int c = 0; c < 2 * CC; ++c) {
      float bvv = gb[(size_t)c * NN + n0 + ml];
#pragma unroll
      for (int v = 0; v < 8; ++v)
        acc[v] += wc[(size_t)(o0 + v + mhf) * (2 * CC) + c] * bvv;
    }
  }
#endif

  int mh = (lane >> 4) * 8;
#pragma unroll
  for (int v = 0; v < 8; ++v) {
    int o = o0 + v + mh;
    out[((size_t)b * OUTC + o) * NN + n0 + ml] = acc[v] + bias[o];
  }
}

// ---------------- launch ----------------
extern "C" void kernel_launch(void* const* d_in, const int* in_sizes, int n_in,
                              void* d_out, int out_size, void* d_ws, size_t ws_size,
                              hipStream_t stream) {
  const float* x    = (const float*)d_in[0];  // (B, C, N)
  const float* W    = (const float*)d_in[1];  // (OUT, 2C, 1, 1)
  const float* bias = (const float*)d_in[2];  // (OUT,)
  float* out = (float*)d_out;                 // (B, OUT, N)

  char*  ws  = (char*)d_ws;
  float* xx  = (float*)ws;                          // B*N floats     (64 KB)
  int*   idx = (int*)(ws + (64 << 10));             // B*N*K ints     (640 KB)
  float* wc  = (float*)(ws + (768 << 10));          // OUT*2C floats  (256 KB)
  float* g   = (float*)(ws + (1 << 20));            // B*2C*N floats  (16 MB)

  xx_kernel<<<(BK * NN) / 256, 256, 0, stream>>>(x, xx);
  knn_topk_kernel<<<BK * (NN / ROWS_PER_WG), 128, 0, stream>>>(x, xx, idx);
  wc_kernel<<<OUTC, 2 * CC, 0, stream>>>(W, wc);
  gather_kernel<<<dim3(BK * CC, NN / 256), 256, 0, stream>>>(x, idx, g);
  out_gemm_kernel<<<(BK * 16 * 256) / 4, 128, 0, stream>>>(wc, g, bias, out);
}